// Sampler_120259084566
// MI455X (gfx1250) — compile-verified
//
#include <hip/hip_runtime.h>
#include <cstdint>
#include <math.h>

#define THREADS 1024
#define NBINS   2048
#define BSHIFT  21
#define CAND    2048
#define CH      1024

typedef unsigned int u32x4 __attribute__((ext_vector_type(4)));
typedef int          i32x8 __attribute__((ext_vector_type(8)));
typedef int          i32x4 __attribute__((ext_vector_type(4)));

// Order-preserving map: float -> uint, ascending (larger float => larger key).
__device__ __forceinline__ unsigned fkey(float x) {
  unsigned u = __float_as_uint(x);
  return (u & 0x80000000u) ? ~u : (u | 0x80000000u);
}

// Issue a TDM 1D tile load (contiguous nelem f32) from global to LDS.
// D# per CDNA5 ISA ch.8; this toolchain's 6-arg builtin (g0,g1,g2,g3,g4,cpol).
__device__ __forceinline__ void tdm_load_1d(const void* gptr, unsigned lds_byte, unsigned nelem) {
  unsigned long long ga = (unsigned long long)(uintptr_t)gptr;
  u32x4 g0;
  g0[0] = 1u;                                             // count=1 (valid), user mode
  g0[1] = lds_byte;                                       // lds_addr
  g0[2] = (unsigned)(ga & 0xFFFFFFFFull);                 // global_addr[31:0]
  g0[3] = (unsigned)((ga >> 32) & 0x1FFFFFFull) | (2u << 30); // global_addr[56:32] | type=2
  i32x8 g1 = {0,0,0,0,0,0,0,0};
  g1[0] = (int)(2u << 16);                                // data_size = 4 bytes, no multicast
  g1[1] = (int)((nelem & 0xFFFFu) << 16);                 // tensor_dim0 [15:0]
  g1[2] = (int)(((nelem >> 16) & 0xFFFFu) | (1u << 16));  // tensor_dim0 hi | tensor_dim1=1
  g1[3] = (int)((nelem & 0xFFFFu) << 16);                 // tile_dim0
  g1[4] = 1;                                              // tile_dim1 = 1
  g1[5] = (int)nelem;                                     // tensor_dim0_stride lo
  i32x4 g2 = {0,0,0,0};
  i32x4 g3 = {0,0,0,0};
  i32x8 g4 = {0,0,0,0,0,0,0,0};
  __builtin_amdgcn_tensor_load_to_lds(g0, g1, g2, g3, g4, 0);
}

__global__ __launch_bounds__(THREADS)
void sampler_kernel(const float* __restrict__ logits,
                    const float* __restrict__ temps,
                    const float* __restrict__ topps,
                    const int*   __restrict__ topks,
                    const float* __restrict__ minps,
                    const float* __restrict__ noise,
                    float* __restrict__ out, int B, int V) {
  __shared__ unsigned h_cnt[NBINS];       // 8 KB (reused as argmax index buf)
  __shared__ float    h_sum[NBINS];       // 8 KB (reused as scan ping-pong)
  __shared__ float    s_a[CAND];          // 8 KB (reduce / candidates)
  __shared__ float    s_b[CAND];          // 8 KB (reduce / scan)
  __shared__ float    s_stage[4 * CH];    // 16 KB (double-buffered TDM staging)
  __shared__ int      s_i[8];
  __shared__ float    s_f[4];
  __shared__ unsigned s_u[2];

  const int row = blockIdx.x;
  const int tid = threadIdx.x;
  const float* xrow = logits + (size_t)row * V;
  const float* qrow = noise  + (size_t)row * V;
  float* out_tok = out;
  float* out_pr  = out + B + (size_t)row * V;
  float* out_lp  = out + B + (size_t)B * V + (size_t)row * V;

  float t = temps[row]; if (t < 1e-5f) t = 1.0f;
  const float invt = 1.0f / t;
  int k = topks[row]; if (k < 1) k = 1;
  const float top_p = topps[row];
  const float min_p = minps[row];

  const int V4 = V >> 2;
  const float4* x4 = (const float4*)xrow;

  // ---- Pass 1: online softmax stats (M1 = row max, S1 = sum exp(x-M1)) ----
  float m = -INFINITY, s = 0.0f;
  for (int i = tid; i < V4; i += THREADS) {
    __builtin_prefetch((const float*)(x4 + i) + 16 * THREADS, 0, 0); // global_prefetch_b8
    float4 c = x4[i];                                     // global_load_b128
    float v0 = c.x * invt, v1 = c.y * invt, v2 = c.z * invt, v3 = c.w * invt;
    float cm = fmaxf(fmaxf(v0, v1), fmaxf(v2, v3));
    float nm = fmaxf(m, cm);
    s = s * __expf(m - nm) + __expf(v0 - nm) + __expf(v1 - nm)
                           + __expf(v2 - nm) + __expf(v3 - nm);
    m = nm;
  }
  for (int i = (V4 << 2) + tid; i < V; i += THREADS) {
    float v = xrow[i] * invt;
    float nm = fmaxf(m, v);
    s = s * __expf(m - nm) + __expf(v - nm);
    m = nm;
  }
  s_a[tid] = m; s_b[tid] = s;
  __syncthreads();
  for (int off = THREADS / 2; off > 0; off >>= 1) {
    if (tid < off) {
      float ma = s_a[tid], mb = s_a[tid + off];
      float nm = fmaxf(ma, mb);
      s_b[tid] = s_b[tid] * __expf(ma - nm) + s_b[tid + off] * __expf(mb - nm);
      s_a[tid] = nm;
    }
    __syncthreads();
  }
  const float M1 = s_a[0];
  const float S1 = s_b[0];
  __syncthreads();

  // ---- Pass 2: histogram (count + exp-mass per key bin), LDS atomics ----
  for (int i = tid; i < NBINS; i += THREADS) { h_cnt[i] = 0u; h_sum[i] = 0.0f; }
  __syncthreads();
  for (int i = tid; i < V4; i += THREADS) {               // L2-resident re-read
    float4 c = x4[i];
    float vv[4] = { c.x * invt, c.y * invt, c.z * invt, c.w * invt };
#pragma unroll
    for (int j = 0; j < 4; ++j) {
      unsigned b = fkey(vv[j]) >> BSHIFT;
      atomicAdd(&h_cnt[b], 1u);
      atomicAdd(&h_sum[b], __expf(vv[j] - M1));
    }
  }
  for (int i = (V4 << 2) + tid; i < V; i += THREADS) {
    float v = xrow[i] * invt;
    unsigned b = fkey(v) >> BSHIFT;
    atomicAdd(&h_cnt[b], 1u);
    atomicAdd(&h_sum[b], __expf(v - M1));
  }
  __syncthreads();

  // ---- Top-down bin scan: kept set is a prefix (m <= k <= 1023) ----
  if (tid == 0) {
    const float pS1s = top_p * S1;
    unsigned C = 0; float P = 0.0f; int bsv = 0;
    for (int b = NBINS - 1; b >= 0; --b) {
      unsigned c = h_cnt[b];
      if (!c) continue;
      if ((C + c <= (unsigned)k) && (P + h_sum[b] <= pS1s)) { C += c; P += h_sum[b]; }
      else { bsv = b; break; }   // guaranteed: V >> k forces a break
    }
    s_i[0] = bsv; s_i[1] = (int)C; s_f[0] = P; s_i[2] = 0; s_i[3] = -1;
  }
  __syncthreads();
  const int bs = s_i[0];
  const unsigned Chi = (unsigned)s_i[1];
  const float Phi = s_f[0];

  // ---- Gather cutoff-bin candidates (<= CAND) ----
  for (int i = tid; i < V4; i += THREADS) {
    float4 c = x4[i];
    float vv[4] = { c.x * invt, c.y * invt, c.z * invt, c.w * invt };
#pragma unroll
    for (int j = 0; j < 4; ++j) {
      if ((int)(fkey(vv[j]) >> BSHIFT) == bs) {
        int pos = atomicAdd(&s_i[2], 1);
        if (pos < CAND) s_a[pos] = vv[j];
      }
    }
  }
  for (int i = (V4 << 2) + tid; i < V; i += THREADS) {
    float v = xrow[i] * invt;
    if ((int)(fkey(v) >> BSHIFT) == bs) {
      int pos = atomicAdd(&s_i[2], 1);
      if (pos < CAND) s_a[pos] = v;
    }
  }
  __syncthreads();
  int cn = s_i[2]; if (cn > CAND) cn = CAND;
  for (int i = tid; i < CAND; i += THREADS) if (i >= cn) s_a[i] = -INFINITY;
  __syncthreads();

  // ---- Bitonic sort descending (2048 elems, LDS) ----
  for (int kk = 2; kk <= CAND; kk <<= 1) {
    for (int j = kk >> 1; j > 0; j >>= 1) {
      for (int i = tid; i < CAND; i += THREADS) {
        int l = i ^ j;
        if (l > i) {
          float a = s_a[i], b = s_a[l];
          bool up = ((i & kk) == 0);
          if (up ? (a < b) : (a > b)) { s_a[i] = b; s_a[l] = a; }
        }
      }
      __syncthreads();
    }
  }

  // ---- Inclusive scan of candidate exp-mass (Hillis-Steele, ping-pong) ----
  for (int i = tid; i < CAND; i += THREADS) s_b[i] = __expf(s_a[i] - M1);
  __syncthreads();
  float* A = s_b; float* Btmp = h_sum;   // histogram no longer needed
  for (int off = 1; off < CAND; off <<= 1) {
    for (int i = tid; i < CAND; i += THREADS)
      Btmp[i] = (i >= off) ? (A[i] + A[i - off]) : A[i];
    __syncthreads();
    float* tmp = A; A = Btmp; Btmp = tmp;
  }
  const float* incl = A;

  // ---- Exact per-element cutoff within the bin -> value threshold tau ----
  const float pS1 = top_p * S1;
  for (int i = tid; i < cn; i += THREADS) {
    float e = __expf(s_a[i] - M1);
    float before = Phi + incl[i] - e;    // prob mass of strictly-higher ranks
    if (((Chi + (unsigned)i) < (unsigned)k) && (before <= pS1))
      atomicMax(&s_i[3], i);
  }
  __syncthreads();
  if (tid == 0) {
    int jl = s_i[3];
    unsigned kthr = (jl >= 0) ? fkey(s_a[jl]) : ((unsigned)(bs + 1) << BSHIFT);
    // min-p after masking reduces to x >= M1 + ln(min_p) (denominator cancels)
    unsigned mthr = fkey(M1 + logf(min_p));
    s_u[0] = (kthr > mthr) ? kthr : mthr;
  }
  __syncthreads();
  const unsigned thr = s_u[0];

  // ---- Pass 3: S3 = sum of kept exp(x - M1) (L2-resident re-read) ----
  float part = 0.0f;
  for (int i = tid; i < V4; i += THREADS) {
    float4 c = x4[i];
    float vv[4] = { c.x * invt, c.y * invt, c.z * invt, c.w * invt };
#pragma unroll
    for (int j = 0; j < 4; ++j)
      if (fkey(vv[j]) >= thr) part += __expf(vv[j] - M1);
  }
  for (int i = (V4 << 2) + tid; i < V; i += THREADS) {
    float v = xrow[i] * invt;
    if (fkey(v) >= thr) part += __expf(v - M1);
  }
  s_a[tid] = part;
  __syncthreads();
  for (int off = THREADS / 2; off > 0; off >>= 1) {
    if (tid < off) s_a[tid] += s_a[tid + off];
    __syncthreads();
  }
  const float S3  = s_a[0];
  const float lS3 = logf(S3);
  __syncthreads();

  // ---- Pass 4: double-buffered TDM stream -> probs/logprobs (NT) + argmax ----
  // TENSORcnt is in-order per issuing wave: with chunk n+1 in flight,
  // s_wait_tensorcnt(2) guarantees chunk n's two loads have landed in LDS.
  float bestv = -1.0f; int besti = 0;
  volatile float* stage = s_stage;
  const int nch = (V + CH - 1) / CH;
  if (tid == 0) {                         // prologue: fill buffer 0 with chunk 0
    int r0 = (V < CH) ? V : CH;
    tdm_load_1d(xrow, (unsigned)(uintptr_t)&s_stage[0],  (unsigned)r0);
    tdm_load_1d(qrow, (unsigned)(uintptr_t)&s_stage[CH], (unsigned)r0);
  }
  for (int c = 0; c < nch; ++c) {
    const int base = c * CH;
    int rem = V - base; if (rem > CH) rem = CH;
    const int cur = c & 1;
    if (c + 1 < nch) {
      if (tid == 0) {                     // prefill the other buffer with chunk c+1
        int nbase = base + CH;
        int nrem = V - nbase; if (nrem > CH) nrem = CH;
        unsigned boff = (unsigned)(2 * CH * (1 - cur));
        tdm_load_1d(xrow + nbase, (unsigned)(uintptr_t)&s_stage[boff],      (unsigned)nrem);
        tdm_load_1d(qrow + nbase, (unsigned)(uintptr_t)&s_stage[boff + CH], (unsigned)nrem);
      }
      if (tid < 32) __builtin_amdgcn_s_wait_tensorcnt(2);
    } else {
      if (tid < 32) __builtin_amdgcn_s_wait_tensorcnt(0);
    }
    __syncthreads();
    if (tid < rem) {
      const int sb = 2 * CH * cur;
      float v = stage[sb + tid] * invt;
      float q = stage[sb + CH + tid];
      bool keep = fkey(v) >= thr;
      float lp = keep ? ((v - M1) - lS3) : -INFINITY;
      float pr = keep ? __expf(lp) : 0.0f;
      __builtin_nontemporal_store(pr, out_pr + base + tid);  // 131MB: keep out of L2
      __builtin_nontemporal_store(lp, out_lp + base + tid);
      float r = pr / q;                                      // exponential-trick sample
      if (r > bestv) { bestv = r; besti = base + tid; }
    }
    __syncthreads();                      // buffer reuse fence before next prefill
  }

  // ---- Argmax reduce (first occurrence on ties) ----
  int* idxb = (int*)h_cnt;
  s_a[tid] = bestv; idxb[tid] = besti;
  __syncthreads();
  for (int off = THREADS / 2; off > 0; off >>= 1) {
    if (tid < off) {
      float v2 = s_a[tid + off]; int i2 = idxb[tid + off];
      if (v2 > s_a[tid] || (v2 == s_a[tid] && i2 < idxb[tid])) { s_a[tid] = v2; idxb[tid] = i2; }
    }
    __syncthreads();
  }
  if (tid == 0) out_tok[row] = (float)idxb[0];
}

extern "C" void kernel_launch(void* const* d_in, const int* in_sizes, int n_in,
                              void* d_out, int out_size, void* d_ws, size_t ws_size,
                              hipStream_t stream) {
  const float* logits = (const float*)d_in[0];
  const float* temps  = (const float*)d_in[1];
  const float* topps  = (const float*)d_in[2];
  const int*   topks  = (const int*)  d_in[3];
  const float* minps  = (const float*)d_in[4];
  const float* noisep = (const float*)d_in[5];
  const int B = in_sizes[1];
  const int V = in_sizes[0] / B;
  sampler_kernel<<<dim3(B), dim3(THREADS), 0, stream>>>(
      logits, temps, topps, topks, minps, noisep, (float*)d_out, B, V);
}